// Angles2Coords_46566035423749
// MI455X (gfx1250) — compile-verified
//
#include <hip/hip_runtime.h>
#include <math.h>

#ifndef M_PI
#define M_PI 3.14159265358979323846
#endif

typedef __attribute__((ext_vector_type(2))) float v2f;
typedef __attribute__((ext_vector_type(8))) float v8f;

#define L_RES    4096
#define NTHREADS 256
#define ITEMS    16     // chain steps handled serially per thread
#define PAD      128    // identity pad = max scan distance
#define NLEVEL   8      // log2(NTHREADS)
#define AFF      16     // floats per scan entry: full 4x4 affine matrix, row-major

struct V3 { float x, y, z; };
struct Rigid { float R[9]; float t[3]; };   // x -> R x + t, R row-major
struct Canon {
    V3 N0, CA0, C0;         // canonical triple
    V3 w0, w1, w2;          // columns of canonical frame rotation Mc
    float ct1, st1;         // cos/sin A_CA_C_N (for N placement)
    float ct2, st2;         // cos/sin A_C_N_CA (for CA placement)
    float ct3, st3;         // cos/sin A_N_CA_C (for C placement)
};

__device__ inline V3 mkv3(float x, float y, float z) { V3 r; r.x=x; r.y=y; r.z=z; return r; }
__device__ inline V3 vsub(V3 a, V3 b) { return mkv3(a.x-b.x, a.y-b.y, a.z-b.z); }
__device__ inline V3 vcross(V3 a, V3 b) {
    return mkv3(a.y*b.z - a.z*b.y, a.z*b.x - a.x*b.z, a.x*b.y - a.y*b.x);
}
__device__ inline V3 vnorm(V3 a) {
    float r = rsqrtf(a.x*a.x + a.y*a.y + a.z*a.z);
    return mkv3(a.x*r, a.y*r, a.z*r);
}

// Frame construction identical to _place_atom's m-matrix: columns [bc, n x bc, n]
__device__ inline void frame3(V3 A, V3 B, V3 C, V3& u0, V3& u1, V3& u2) {
    u0 = vnorm(vsub(C, B));                 // bc
    u2 = vnorm(vcross(vsub(B, A), u0));     // n
    u1 = vcross(u2, u0);                    // n x bc
}

__device__ inline V3 place_atom(V3 A, V3 B, V3 C, float Rb, float ct, float st, float phi) {
    V3 bc = vnorm(vsub(C, B));
    V3 n  = vnorm(vcross(vsub(B, A), bc));
    V3 m1 = vcross(n, bc);
    float sp, cp;
    sincosf(phi, &sp, &cp);
    float d0 = -Rb * ct;
    float d1 =  Rb * st * cp;
    float d2 =  Rb * st * sp;
    return mkv3(C.x + bc.x*d0 + m1.x*d1 + n.x*d2,
                C.y + bc.y*d0 + m1.y*d1 + n.y*d2,
                C.z + bc.z*d0 + m1.z*d1 + n.z*d2);
}

__device__ inline Rigid rigid_identity() {
    Rigid o;
    #pragma unroll
    for (int i = 0; i < 9; ++i) o.R[i] = (i % 4 == 0) ? 1.0f : 0.0f;
    o.t[0] = o.t[1] = o.t[2] = 0.0f;
    return o;
}

// (F o G)(x) = F(G(x)) : earlier transform on the LEFT
__device__ inline Rigid rcompose(const Rigid& F, const Rigid& G) {
    Rigid o;
    #pragma unroll
    for (int r = 0; r < 3; ++r) {
        #pragma unroll
        for (int c = 0; c < 3; ++c)
            o.R[3*r+c] = F.R[3*r+0]*G.R[0+c] + F.R[3*r+1]*G.R[3+c] + F.R[3*r+2]*G.R[6+c];
        o.t[r] = F.R[3*r+0]*G.t[0] + F.R[3*r+1]*G.t[1] + F.R[3*r+2]*G.t[2] + F.t[r];
    }
    return o;
}

__device__ inline V3 rapply(const Rigid& Q, V3 p) {
    return mkv3(Q.R[0]*p.x + Q.R[1]*p.y + Q.R[2]*p.z + Q.t[0],
                Q.R[3]*p.x + Q.R[4]*p.y + Q.R[5]*p.z + Q.t[1],
                Q.R[6]*p.x + Q.R[7]*p.y + Q.R[8]*p.z + Q.t[2]);
}

// Store rigid as rows 0..2 of a 4x4 affine matrix (row 3 is pre-initialized [0,0,0,1])
__device__ inline void rstore16(float* p, const Rigid& r) {
    #pragma unroll
    for (int rr = 0; rr < 3; ++rr) {
        p[4*rr + 0] = r.R[3*rr + 0];
        p[4*rr + 1] = r.R[3*rr + 1];
        p[4*rr + 2] = r.R[3*rr + 2];
        p[4*rr + 3] = r.t[rr];
    }
}
__device__ inline Rigid rload16(const float* p) {
    Rigid r;
    #pragma unroll
    for (int rr = 0; rr < 3; ++rr) {
        r.R[3*rr + 0] = p[4*rr + 0];
        r.R[3*rr + 1] = p[4*rr + 1];
        r.R[3*rr + 2] = p[4*rr + 2];
        r.t[rr]       = p[4*rr + 3];
    }
    return r;
}

__device__ inline Canon make_canon() {
    Canon c;
    const float A_N_CA_C = (float)(111.2 * M_PI / 180.0);
    const float A_CA_C_N = (float)(116.2 * M_PI / 180.0);
    const float A_C_N_CA = (float)(121.7 * M_PI / 180.0);
    c.N0  = mkv3(0.f, 0.f, 0.f);
    c.CA0 = mkv3(1.458f, 0.f, 0.f);
    float a = (float)M_PI - A_N_CA_C;
    c.C0  = mkv3(1.458f + 1.525f * cosf(a), 1.525f * sinf(a), 0.f);
    c.ct1 = cosf(A_CA_C_N); c.st1 = sinf(A_CA_C_N);
    c.ct2 = cosf(A_C_N_CA); c.st2 = sinf(A_C_N_CA);
    c.ct3 = cosf(A_N_CA_C); c.st3 = sinf(A_N_CA_C);
    frame3(c.N0, c.CA0, c.C0, c.w0, c.w1, c.w2);
    return c;
}

// Per-step local triple + step rigid transform G_j = Phi(local triple) o Phi(canon)^-1
__device__ inline void computeGL(int j, const float* __restrict__ ang, const Canon& cn,
                                 Rigid& G, V3& Nn, V3& CAn, V3& Cn) {
    const float psi = ang[1 * L_RES + j];       // psi[:-1][j]
    const float om  = ang[2 * L_RES + j + 1];   // omega[1:][j]
    const float ph  = ang[0 * L_RES + j + 1];   // phi[1:][j]
    Nn  = place_atom(cn.N0,  cn.CA0, cn.C0, 1.329f, cn.ct1, cn.st1, psi);
    CAn = place_atom(cn.CA0, cn.C0,  Nn,    1.458f, cn.ct2, cn.st2, om);
    Cn  = place_atom(cn.C0,  Nn,    CAn,   1.525f, cn.ct3, cn.st3, ph);
    V3 u0, u1, u2;
    frame3(Nn, CAn, Cn, u0, u1, u2);
    // Rg = u0*w0^T + u1*w1^T + u2*w2^T   (m_local * Mc^T)
    G.R[0] = u0.x*cn.w0.x + u1.x*cn.w1.x + u2.x*cn.w2.x;
    G.R[1] = u0.x*cn.w0.y + u1.x*cn.w1.y + u2.x*cn.w2.y;
    G.R[2] = u0.x*cn.w0.z + u1.x*cn.w1.z + u2.x*cn.w2.z;
    G.R[3] = u0.y*cn.w0.x + u1.y*cn.w1.x + u2.y*cn.w2.x;
    G.R[4] = u0.y*cn.w0.y + u1.y*cn.w1.y + u2.y*cn.w2.y;
    G.R[5] = u0.y*cn.w0.z + u1.y*cn.w1.z + u2.y*cn.w2.z;
    G.R[6] = u0.z*cn.w0.x + u1.z*cn.w1.x + u2.z*cn.w2.x;
    G.R[7] = u0.z*cn.w0.y + u1.z*cn.w1.y + u2.z*cn.w2.y;
    G.R[8] = u0.z*cn.w0.z + u1.z*cn.w1.z + u2.z*cn.w2.z;
    // t = Cn - Rg * C0
    G.t[0] = Cn.x - (G.R[0]*cn.C0.x + G.R[1]*cn.C0.y + G.R[2]*cn.C0.z);
    G.t[1] = Cn.y - (G.R[3]*cn.C0.x + G.R[4]*cn.C0.y + G.R[5]*cn.C0.z);
    G.t[2] = Cn.z - (G.R[6]*cn.C0.x + G.R[7]*cn.C0.y + G.R[8]*cn.C0.z);
}

__global__ __launch_bounds__(NTHREADS)
void nerf_wmma_scan_kernel(const float* __restrict__ ang, float* __restrict__ out) {
    // double-buffered scan array: [identity pad (PAD) | NTHREADS aggregates] x 4x4 affine
    __shared__ float Sb[2][(PAD + NTHREADS) * AFF];

    const int t = threadIdx.x;
    const Canon cn = make_canon();

    // prefetch angle rows this thread consumes (global_prefetch_b8)
    __builtin_prefetch(ang + 1 * L_RES + t * ITEMS, 0, 0);
    __builtin_prefetch(ang + 2 * L_RES + t * ITEMS, 0, 0);

    // ---- identity pad in both buffers ----
    if (t < PAD) {
        #pragma unroll
        for (int e = 0; e < AFF; ++e) {
            float v = (e % 5 == 0) ? 1.0f : 0.0f;   // indices 0,5,10,15 -> diagonal
            Sb[0][t * AFF + e] = v;
            Sb[1][t * AFF + e] = v;
        }
    }
    // ---- row 3 = [0,0,0,1] of own entry, both buffers (invariant under affine product) ----
    {
        const int b = (PAD + t) * AFF + 12;
        Sb[0][b+0] = 0.f; Sb[0][b+1] = 0.f; Sb[0][b+2] = 0.f; Sb[0][b+3] = 1.f;
        Sb[1][b+0] = 0.f; Sb[1][b+1] = 0.f; Sb[1][b+2] = 0.f; Sb[1][b+3] = 1.f;
    }

    // ---- Phase A: per-thread serial aggregate of 16 step transforms ----
    Rigid agg = rigid_identity();
    #pragma unroll 4
    for (int k = 0; k < ITEMS; ++k) {
        int j = t * ITEMS + k;
        if (j < L_RES - 1) {
            Rigid G; V3 a, b, c;
            computeGL(j, ang, cn, G, a, b, c);
            agg = rcompose(agg, G);
        }
    }
    rstore16(&Sb[0][(PAD + t) * AFF], agg);

    // ---- Phase B: Hillis-Steele scan over 256 rigid transforms.
    // Compose = 4x4 affine matmul; 4 composes batched into one
    // V_WMMA_F32_16X16X4_F32 (K=4 == inner dim). Diagonal D blocks are kept.
    const int wave = t >> 5;
    const int lane = t & 31;
    const int h  = lane >> 4;       // lane half
    const int mm = lane & 15;
    const int p  = mm >> 2;         // 4x4 block index (A: row block, B: col block)
    const int rc = mm & 3;          // A: row within block / B: col within block

    // writeback role: which diagonal block / column this lane owns
    int pp = -1, vb = 0, cc = 0;
    if (lane < 8)        { pp = lane >> 2;            vb = 4 * pp;       cc = lane & 3; }
    else if (lane >= 24) { int n = lane - 16; pp = n >> 2; vb = 4 * (pp - 2); cc = n & 3; }

    for (int lvl = 0; lvl < NLEVEL; ++lvl) {
        __syncthreads();
        const int d = 1 << lvl;
        const float* src = Sb[lvl & 1];
        float* dst = Sb[(lvl + 1) & 1];

        for (int q = 0; q < 8; ++q) {
            const int gbase = wave * 32 + 4 * q;   // 4 scan entries per WMMA
            const int ia = (PAD + gbase + p - d) * AFF;  // A = earlier prefix S[t-d]
            const int ib = (PAD + gbase + p) * AFF;      // B = current S[t]

            // A (16x4): lane half h, VGPR v -> K = 2h+v; row M = mm -> block p, row rc
            // contiguous pair -> single 8-byte LDS load (even float offset)
            v2f a = *(const v2f*)(src + ia + 4 * rc + 2 * h);
            // B (4x16): VGPR v -> K = 2h+v; col N = mm -> block p, col rc
            v2f b;
            b.x = src[ib + 8 * h + rc];          // row K=2h,   col rc
            b.y = src[ib + 8 * h + 4 + rc];      // row K=2h+1, col rc

            v8f cz = {0.f, 0.f, 0.f, 0.f, 0.f, 0.f, 0.f, 0.f};
            v8f dd = __builtin_amdgcn_wmma_f32_16x16x4_f32(
                false, a, false, b, (short)0, cz, false, false);

            if (pp >= 0) {
                // D: VGPR v holds row (v + 8h), col = lane&15 -> rows 4pp..4pp+2 of block pp
                float e0, e1, e2;
                if (vb == 0) { e0 = dd[0]; e1 = dd[1]; e2 = dd[2]; }
                else         { e0 = dd[4]; e1 = dd[5]; e2 = dd[6]; }
                float* o = dst + (PAD + gbase + pp) * AFF;
                o[0 + cc] = e0;   // row 0 of block, column cc (cc==3 -> translation)
                o[4 + cc] = e1;   // row 1
                o[8 + cc] = e2;   // row 2
            }
        }
    }
    __syncthreads();

    // after 8 levels (even), inclusive scan is back in buffer 0
    const float* fb = Sb[0];

    // ---- Phase C: exclusive prefix + emit coordinates ----
    // t==0 reads pad entry (identity) automatically
    Rigid Q = rload16(&fb[(PAD + t - 1) * AFF]);
    for (int k = 0; k < ITEMS; ++k) {
        int j = t * ITEMS + k;
        if (j >= L_RES - 1) break;
        Rigid G; V3 Nn, CAn, Cn;
        computeGL(j, ang, cn, G, Nn, CAn, Cn);
        V3 gN = rapply(Q, Nn), gCA = rapply(Q, CAn), gC = rapply(Q, Cn);
        float* o = out + 9 * (j + 1);
        o[0] = gN.x;  o[1] = gN.y;  o[2] = gN.z;
        o[3] = gCA.x; o[4] = gCA.y; o[5] = gCA.z;
        o[6] = gC.x;  o[7] = gC.y;  o[8] = gC.z;
        Q = rcompose(Q, G);
    }

    if (t == 0) {  // residue 0 = canonical triple
        out[0] = 0.f;      out[1] = 0.f;      out[2] = 0.f;
        out[3] = cn.CA0.x; out[4] = cn.CA0.y; out[5] = cn.CA0.z;
        out[6] = cn.C0.x;  out[7] = cn.C0.y;  out[8] = cn.C0.z;
    }
}

extern "C" void kernel_launch(void* const* d_in, const int* in_sizes, int n_in,
                              void* d_out, int out_size, void* d_ws, size_t ws_size,
                              hipStream_t stream) {
    (void)in_sizes; (void)n_in; (void)out_size; (void)d_ws; (void)ws_size;
    const float* ang = (const float*)d_in[0];
    float* out = (float*)d_out;
    nerf_wmma_scan_kernel<<<dim3(1), dim3(NTHREADS), 0, stream>>>(ang, out);
}